// VarianceAdaptor_39084202394284
// MI455X (gfx1250) — compile-verified
//
#include <hip/hip_runtime.h>
#include <hip/hip_bf16.h>
#include <math.h>

typedef __attribute__((ext_vector_type(2))) float v2f;
typedef __attribute__((ext_vector_type(8))) float v8f;
typedef __attribute__((ext_vector_type(4))) int   v4i;

#define CIN 256
#define COUT 256
#define KTOT 768          // KS * CIN = 3 * 256
#define LN_EPS_F 1e-5f
#define BS2 544           // pair-row stride (272 float2); 544 % 64 == 32 -> halves on disjoint banks

#if __has_builtin(__builtin_amdgcn_global_load_async_to_lds_b128)
#define USE_ASYNC_LDS 1
typedef __attribute__((address_space(1))) v4i* g4p_t;   // global int4*
typedef __attribute__((address_space(3))) v4i* l4p_t;   // LDS int4*
__device__ __forceinline__ void async_copy16(const float* src, float* dst) {
    __builtin_amdgcn_global_load_async_to_lds_b128((g4p_t)src, (l4p_t)dst, 0, 0);
}
__device__ __forceinline__ void async_wait0() {
#if __has_builtin(__builtin_amdgcn_s_wait_asynccnt)
    __builtin_amdgcn_s_wait_asynccnt(0);
#else
    asm volatile("s_wait_asynccnt 0x0" ::: "memory");
#endif
}
#else
#define USE_ASYNC_LDS 0
#endif

// ---------------------------------------------------------------------------
// Fused Conv1d(KS=3, same) + bias + ReLU + LayerNorm [+ optional head dot]
// GEMM view: (B*S, 768) x (768, 256). One block computes a 32x256 output tile.
// 256 threads = 8 waves; each wave owns a 32(M) x 32(N) tile via
// 4 x V_WMMA_F32_16X16X4_F32 accumulators. B (weights) arrive pre-packed
// K-pair-interleaved (384, 512) so staging is a straight copy -> async-to-LDS.
// ---------------------------------------------------------------------------
__global__ __launch_bounds__(256)
void conv_block_wmma(const float* __restrict__ X, int S,
                     const float* __restrict__ Wp,     // (384, 512) pair-interleaved
                     const float* __restrict__ bias,   // (256)
                     const float* __restrict__ gamma,  // (256)
                     const float* __restrict__ beta,   // (256)
                     float* __restrict__ Y,            // (rows, 256) or null
                     const float* __restrict__ wh,     // (256) or null
                     const float* __restrict__ bh,     // (1) or null
                     float* __restrict__ head)         // (rows) or null
{
    __shared__ float As[32][36];        // 32 rows x 32 K, stride 36 (144B, 16B aligned)
    __shared__ float Bs[16 * BS2];      // pair-interleaved B tile; reused as Ys[32][256]
    __shared__ float red[32][8][2];
    __shared__ float mu[32], rstd[32];

    const int tid  = threadIdx.x;
    const int wave = tid >> 5;
    const int lane = tid & 31;
    const int m0   = blockIdx.x * 32;   // 32 | S so a block never crosses batch
    const int b    = m0 / S;
    const int t0   = m0 % S;

    const int l15   = lane & 15;
    const int half2 = (lane >> 4) << 1; // 0 or 2 (K pair selector)
    const int nbase = wave << 5;        // 32 N columns per wave

    // A-tile staging coordinates: one float4 per thread (32 rows x 8 x float4)
    const int ar  = tid >> 3;           // 0..31
    const int ac4 = (tid & 7) << 2;     // 0..28 step 4
    const bool interior = (t0 > 0) && (t0 + 32 < S);  // block-uniform

    v8f acc[2][2] = {};

    for (int kk0 = 0; kk0 < KTOT; kk0 += 32) {
        // ---- stage A tile (im2col; interior blocks need no zero padding)
        {
            int kk  = kk0 + ac4;        // 32 | 256 so one tap per tile-row-chunk
            int tap = kk >> 8;
            int ci  = kk & 255;
            int tt  = t0 + ar + tap - 1;
            const float* src = &X[((size_t)b * S + tt) * CIN + ci];
            float* dst = &As[ar][ac4];
#if USE_ASYNC_LDS
            if (interior) {
                async_copy16(src, dst);
            } else
#endif
            {
                float4 v = make_float4(0.f, 0.f, 0.f, 0.f);
                if (tt >= 0 && tt < S) v = *(const float4*)src;
                *(float4*)dst = v;
            }
        }
        // ---- stage B tile: straight copy of 16 pair-rows x 512 floats
        {
            const float* wp = Wp + (size_t)(kk0 >> 1) * 512;
            #pragma unroll
            for (int it = tid; it < 16 * 128; it += 256) {   // 8 x b128 per thread
                int row = it >> 7;            // 0..15
                int c4  = (it & 127) << 2;    // 0..508 step 4
                const float* src = wp + row * 512 + c4;
                float* dst = &Bs[row * BS2 + c4];
#if USE_ASYNC_LDS
                async_copy16(src, dst);
#else
                *(float4*)dst = *(const float4*)src;
#endif
            }
        }
#if USE_ASYNC_LDS
        async_wait0();
#endif
        __syncthreads();

        #pragma unroll
        for (int k4 = 0; k4 < 32; k4 += 4) {
            const int p = (k4 + half2) >> 1;        // pair row for this lane half
            v2f a0, a1, b0, b1;
            a0.x = As[l15][k4 + half2];
            a0.y = As[l15][k4 + half2 + 1];
            a1.x = As[16 + l15][k4 + half2];
            a1.y = As[16 + l15][k4 + half2 + 1];
            b0 = *(const v2f*)&Bs[p * BS2 + 2 * (nbase + l15)];
            b1 = *(const v2f*)&Bs[p * BS2 + 2 * (nbase + 16 + l15)];
            acc[0][0] = __builtin_amdgcn_wmma_f32_16x16x4_f32(false, a0, false, b0, (short)0, acc[0][0], false, false);
            acc[0][1] = __builtin_amdgcn_wmma_f32_16x16x4_f32(false, a0, false, b1, (short)0, acc[0][1], false, false);
            acc[1][0] = __builtin_amdgcn_wmma_f32_16x16x4_f32(false, a1, false, b0, (short)0, acc[1][0], false, false);
            acc[1][1] = __builtin_amdgcn_wmma_f32_16x16x4_f32(false, a1, false, b1, (short)0, acc[1][1], false, false);
        }
        __syncthreads();
    }

    // ---- bias + ReLU into LDS (reuse Bs as Ys[32][256])
    {
        int rhalf = (lane >> 4) << 3;               // +0 or +8
        #pragma unroll
        for (int ni = 0; ni < 2; ++ni) {
            int col = nbase + (ni << 4) + l15;
            float bv = bias[col];
            #pragma unroll
            for (int mi = 0; mi < 2; ++mi) {
                #pragma unroll
                for (int r = 0; r < 8; ++r) {
                    int row = (mi << 4) + rhalf + r;
                    float v = acc[mi][ni][r] + bv;
                    Bs[row * 256 + col] = v > 0.f ? v : 0.f;
                }
            }
        }
    }
    __syncthreads();

    // ---- LayerNorm over the 256 channels of each of the 32 rows
    const int r   = tid >> 3;   // 0..31
    const int seg = tid & 7;    // 8 segments of 32 channels
    float s = 0.f, ss = 0.f;
    #pragma unroll
    for (int j = 0; j < 32; ++j) {
        float v = Bs[r * 256 + seg * 32 + j];
        s += v; ss += v * v;
    }
    red[r][seg][0] = s;
    red[r][seg][1] = ss;
    __syncthreads();
    if (tid < 32) {
        float s2 = 0.f, ss2 = 0.f;
        #pragma unroll
        for (int j = 0; j < 8; ++j) { s2 += red[tid][j][0]; ss2 += red[tid][j][1]; }
        float m   = s2 * (1.f / 256.f);
        float var = ss2 * (1.f / 256.f) - m * m;    // biased var (matches jnp.var)
        mu[tid]   = m;
        rstd[tid] = rsqrtf(var + LN_EPS_F);
    }
    __syncthreads();

    const int m = m0 + r;
    const float mym = mu[r], myr = rstd[r];
    float hsum = 0.f;
    for (int j = 0; j < 32; ++j) {
        int c = seg * 32 + j;
        float v = (Bs[r * 256 + c] - mym) * myr * gamma[c] + beta[c];
        if (Y)  Y[(size_t)m * COUT + c] = v;
        if (wh) hsum += v * wh[c];
    }
    if (wh) {   // uniform condition across the block: barrier is safe
        red[r][seg][0] = hsum;
        __syncthreads();
        if (seg == 0) {
            float hs = 0.f;
            #pragma unroll
            for (int j = 0; j < 8; ++j) hs += red[r][j][0];
            head[m] = hs + bh[0];
        }
    }
}

// Pre-pack conv weight (768, 256) -> K-pair-interleaved (384, 512):
// Wp[p][2n+0] = W[2p][n], Wp[p][2n+1] = W[2p+1][n]
__global__ __launch_bounds__(256)
void prepack_w(const float* __restrict__ W, float* __restrict__ Wp)
{
    int i = blockIdx.x * 256 + threadIdx.x;   // (p, n)
    int p = i >> 8, n = i & 255;
    float e = W[(size_t)(2 * p) * COUT + n];
    float o = W[(size_t)(2 * p + 1) * COUT + n];
    *(float2*)&Wp[(size_t)p * 512 + 2 * n] = make_float2(e, o);
}

// ---------------------------------------------------------------------------
// Length regulator: per batch, inclusive scan of durations, then each phoneme
// scatters its index over its frame range.  idx[t] = searchsorted(csum, t, 'right')
// ---------------------------------------------------------------------------
__global__ __launch_bounds__(512)
void length_regulator(const int* __restrict__ dur, int* __restrict__ idx,
                      int S, int OUT)
{
    __shared__ int cs[512];
    const int b   = blockIdx.x;
    const int tid = threadIdx.x;     // blockDim == S == 512
    cs[tid] = dur[b * S + tid];
    __syncthreads();
    for (int off = 1; off < 512; off <<= 1) {
        int v = (tid >= off) ? cs[tid - off] : 0;
        __syncthreads();
        cs[tid] += v;
        __syncthreads();
    }
    int total = cs[S - 1];
    for (int t = tid; t < OUT; t += 512)
        if (t >= total) idx[b * OUT + t] = S;       // invalid -> zero rows
    __syncthreads();
    int lo = (tid == 0) ? 0 : cs[tid - 1];
    int hi = cs[tid];
    for (int t = lo; t < hi && t < OUT; ++t)
        idx[b * OUT + t] = tid;
}

// aligned[b,t,:] = (idx < S) ? X[b, idx, :] : 0
__global__ __launch_bounds__(256)
void gather_aligned(const float* __restrict__ X, const int* __restrict__ idx,
                    float* __restrict__ A, int S, int OUT)
{
    int row = blockIdx.x;            // b*OUT + t
    int b   = row / OUT;
    int j   = idx[row];
    float v = 0.f;
    if (j < S) v = X[((size_t)b * S + j) * 256 + threadIdx.x];
    A[(size_t)row * 256 + threadIdx.x] = v;
}

// searchsorted(bounds, x, side='left') with analytic bounds, 255 entries
__device__ __forceinline__ int bucket_search(float x, bool logmode)
{
    int lo = 0, hi = 255;
    const float lk = logf(100.f) / 254.f;
    while (lo < hi) {
        int mid = (lo + hi) >> 1;    // mid in [0, 254]
        float bnd = logmode ? expf(lk * (float)mid)
                            : (1.f + (99.f / 254.f) * (float)mid);
        if (bnd >= x) hi = mid; else lo = mid + 1;
    }
    return lo;
}

__global__ __launch_bounds__(256)
void fuse_out_emb(const float* __restrict__ A,
                  const float* __restrict__ pitch,
                  const float* __restrict__ energy,
                  const float* __restrict__ pemb,
                  const float* __restrict__ eemb,
                  float* __restrict__ out)
{
    int row = blockIdx.x;
    int c   = threadIdx.x;
    int pi  = bucket_search(pitch[row],  true);
    int ei  = bucket_search(energy[row], false);
    size_t o = (size_t)row * 256 + c;
    out[o] = A[o] + pemb[pi * 256 + c] + eemb[ei * 256 + c];
}

// ---------------------------------------------------------------------------
struct Pred {
    const float *w1, *b1, *g1, *be1, *w2, *b2, *g2, *be2, *wh, *bh;
};

extern "C" void kernel_launch(void* const* d_in, const int* in_sizes, int n_in,
                              void* d_out, int out_size, void* d_ws, size_t ws_size,
                              hipStream_t stream)
{
    (void)n_in; (void)out_size; (void)ws_size;
    const int B = 32, S = 512, OUT = 2304, D = 256;
    const int WPN = (KTOT / 2) * 512;              // 196608 floats per packed weight

    const float* X      = (const float*)d_in[0];   // (B,S,D)
    const int*   dur_in = (const int*)d_in[1];     // (B,S)
    // d_in[2] = out_len scalar (fixed 2304)

    auto F = [&](int i) { return (const float*)d_in[i]; };
    Pred dur, pitch, energy;
    const float *pemb, *eemb;
    if (in_sizes[3] == 196608) {
        // insertion order: {w1,b1,g1,be1,w2,b2,g2,be2,wh,bh} x {dur,pitch,energy}, pitch_emb, energy_emb
        auto load = [&](int o, Pred& p) {
            p.w1=F(o+0); p.b1=F(o+1); p.g1=F(o+2); p.be1=F(o+3); p.w2=F(o+4);
            p.b2=F(o+5); p.g2=F(o+6); p.be2=F(o+7); p.wh=F(o+8); p.bh=F(o+9);
        };
        load(3, dur); load(13, pitch); load(23, energy);
        pemb = F(33); eemb = F(34);
    } else {
        // jax tree-sorted: {b1,b2,be1,be2,bh,g1,g2,w1,w2,wh} x {dur, energy}, energy_emb, pitch, pitch_emb
        auto load = [&](int o, Pred& p) {
            p.b1=F(o+0); p.b2=F(o+1); p.be1=F(o+2); p.be2=F(o+3); p.bh=F(o+4);
            p.g1=F(o+5); p.g2=F(o+6); p.w1=F(o+7); p.w2=F(o+8); p.wh=F(o+9);
        };
        load(3, dur); load(13, energy); eemb = F(23);
        load(24, pitch); pemb = F(34);
    }

    float* out      = (float*)d_out;
    float* o_dur    = out;                              // (B,S)
    float* o_pitch  = o_dur   + (size_t)B * S;          // (B,OUT)
    float* o_energy = o_pitch + (size_t)B * OUT;        // (B,OUT)
    float* o_emb    = o_energy + (size_t)B * OUT;       // (B,OUT,D)

    float* aligned = (float*)d_ws;                      // 75.5 MB
    float* y1      = aligned + (size_t)B * OUT * D;     // 75.5 MB
    int*   idxbuf  = (int*)(y1 + (size_t)B * OUT * D);  // 0.3 MB
    float* wp      = (float*)(idxbuf + (size_t)B * OUT);// 6 x 768 KB packed weights
    float* wp_d1 = wp;            float* wp_d2 = wp + 1 * (size_t)WPN;
    float* wp_p1 = wp + 2 * (size_t)WPN; float* wp_p2 = wp + 3 * (size_t)WPN;
    float* wp_e1 = wp + 4 * (size_t)WPN; float* wp_e2 = wp + 5 * (size_t)WPN;

    // pre-pack all six conv weights into K-pair-interleaved layout
    const int PG = (KTOT / 2) * COUT / 256;             // 384 blocks
    prepack_w<<<PG, 256, 0, stream>>>(dur.w1,    wp_d1);
    prepack_w<<<PG, 256, 0, stream>>>(dur.w2,    wp_d2);
    prepack_w<<<PG, 256, 0, stream>>>(pitch.w1,  wp_p1);
    prepack_w<<<PG, 256, 0, stream>>>(pitch.w2,  wp_p2);
    prepack_w<<<PG, 256, 0, stream>>>(energy.w1, wp_e1);
    prepack_w<<<PG, 256, 0, stream>>>(energy.w2, wp_e2);

    // duration predictor on phoneme sequence
    conv_block_wmma<<<B * S / 32, 256, 0, stream>>>(X, S, wp_d1, dur.b1, dur.g1, dur.be1,
                                                    y1, nullptr, nullptr, nullptr);
    conv_block_wmma<<<B * S / 32, 256, 0, stream>>>(y1, S, wp_d2, dur.b2, dur.g2, dur.be2,
                                                    nullptr, dur.wh, dur.bh, o_dur);
    // expand to frames
    length_regulator<<<B, 512, 0, stream>>>(dur_in, idxbuf, S, OUT);
    gather_aligned<<<B * OUT, 256, 0, stream>>>(X, idxbuf, aligned, S, OUT);
    // pitch predictor
    conv_block_wmma<<<B * OUT / 32, 256, 0, stream>>>(aligned, OUT, wp_p1, pitch.b1, pitch.g1, pitch.be1,
                                                      y1, nullptr, nullptr, nullptr);
    conv_block_wmma<<<B * OUT / 32, 256, 0, stream>>>(y1, OUT, wp_p2, pitch.b2, pitch.g2, pitch.be2,
                                                      nullptr, pitch.wh, pitch.bh, o_pitch);
    // energy predictor
    conv_block_wmma<<<B * OUT / 32, 256, 0, stream>>>(aligned, OUT, wp_e1, energy.b1, energy.g1, energy.be1,
                                                      y1, nullptr, nullptr, nullptr);
    conv_block_wmma<<<B * OUT / 32, 256, 0, stream>>>(y1, OUT, wp_e2, energy.b2, energy.g2, energy.be2,
                                                      nullptr, energy.wh, energy.bh, o_energy);
    // bucketize + embedding add
    fuse_out_emb<<<B * OUT, 256, 0, stream>>>(aligned, o_pitch, o_energy, pemb, eemb, o_emb);
}